// Involution_2534030705179
// MI455X (gfx1250) — compile-verified
//
#include <hip/hip_runtime.h>

// Involution layer, fp32, MI455X (gfx1250, wave32, WMMA).
//   t    = relu(BN(x @ w1 + b1))          GEMM1: M=8192 K=256 N=64   (v_wmma_f32_16x16x4_f32)
//   kern = t @ w2 + b2                    GEMM2: M=8192 K=64  N=784  (v_wmma_f32_16x16x4_f32)
//   out[p,c] = sum_kk kern[p,kk,c%16] * xpad[p+off(kk), c]   (gather, LDS-cached kern)

#define K_SZ   7
#define KK     49
#define G_NUM  16
#define C_CH   256
#define CR     64            // C / reduction_ratio
#define NK     (KK * G_NUM)  // 784
#define HW     64
#define BATCH  2
#define M_TOT  (BATCH * HW * HW)  // 8192
#define BN_EPS 1e-3f

typedef __attribute__((ext_vector_type(2))) float v2f;
typedef __attribute__((ext_vector_type(8))) float v8f;

// ---------------------------------------------------------------------------
// GEMM1: t[m, n] = relu( BN( sum_k x[m,k] * w1[k,n] + b1[n] ) ), n < 64
// One 16x16 tile per wave; K stepped by 4 with V_WMMA_F32_16X16X4_F32.
// ---------------------------------------------------------------------------
__global__ __launch_bounds__(256) void gemm1_bn_relu(
    const float* __restrict__ x,  const float* __restrict__ w1,
    const float* __restrict__ b1, const float* __restrict__ gamma,
    const float* __restrict__ beta, const float* __restrict__ mu,
    const float* __restrict__ var, float* __restrict__ t)
{
    const int wave  = threadIdx.x >> 5;
    const int ln    = threadIdx.x & 31;
    const int tile  = blockIdx.x * 8 + wave;      // 2048 tiles total
    const int m0    = (tile >> 2) << 4;           // 512 m-tiles
    const int n0    = (tile & 3)  << 4;           // 4 n-tiles
    const int laneM = ln & 15;                    // A: row, B/D: col
    const int kHalf = ln >> 4;                    // selects K pair {0,1} vs {2,3}

    // A(16x4) per-lane: a.x = x[m0+laneM][k + 2*kHalf], a.y = next k
    const float* arow = x + (size_t)(m0 + laneM) * C_CH + 2 * kHalf;
    // B(4x16) per-lane: b.x = w1[k + 2*kHalf][n0+laneM], b.y = next k
    const float* bcol = w1 + (size_t)(2 * kHalf) * CR + (n0 + laneM);

    v8f acc = {};
#pragma unroll 8
    for (int k = 0; k < C_CH; k += 4) {
        v2f a, b;
        a.x = arow[k];
        a.y = arow[k + 1];
        b.x = bcol[(size_t)k * CR];
        b.y = bcol[(size_t)(k + 1) * CR];
        acc = __builtin_amdgcn_wmma_f32_16x16x4_f32(
            false, a, false, b, (short)0, acc, false, false);
    }

    // Fused epilogue: +b1, BN(inference), ReLU.  Per-column affine.
    const int   col   = n0 + laneM;
    const float scale = gamma[col] * rsqrtf(var[col] + BN_EPS);
    const float shift = beta[col] - mu[col] * scale;
    const float b1c   = b1[col];
    const int   rowB  = m0 + (kHalf << 3);  // D: VGPR r -> row r (lanes 0-15) / r+8
#pragma unroll
    for (int r = 0; r < 8; ++r) {
        float v = fmaf(acc[r] + b1c, scale, shift);
        t[(size_t)(rowB + r) * CR + col] = fmaxf(v, 0.0f);
    }
}

// ---------------------------------------------------------------------------
// GEMM2: kern[m, n] = sum_k t[m,k] * w2[k,n] + b2[n], n < 784
// ---------------------------------------------------------------------------
__global__ __launch_bounds__(256) void gemm2_kerngen(
    const float* __restrict__ t, const float* __restrict__ w2,
    const float* __restrict__ b2, float* __restrict__ kern)
{
    const int wave = threadIdx.x >> 5;
    const int ln   = threadIdx.x & 31;
    const int tile = blockIdx.x * 8 + wave;       // 512 * 49 = 25088 tiles
    if (tile >= (M_TOT / 16) * (NK / 16)) return;
    const int mt = tile / (NK / 16);
    const int nt = tile % (NK / 16);
    const int m0 = mt << 4, n0 = nt << 4;
    const int laneM = ln & 15;
    const int kHalf = ln >> 4;

    const float* arow = t  + (size_t)(m0 + laneM) * CR + 2 * kHalf;
    const float* bcol = w2 + (size_t)(2 * kHalf) * NK + (n0 + laneM);

    v8f acc = {};
#pragma unroll
    for (int k = 0; k < CR; k += 4) {
        v2f a, b;
        a.x = arow[k];
        a.y = arow[k + 1];
        b.x = bcol[(size_t)k * NK];
        b.y = bcol[(size_t)(k + 1) * NK];
        acc = __builtin_amdgcn_wmma_f32_16x16x4_f32(
            false, a, false, b, (short)0, acc, false, false);
    }

    const int   col  = n0 + laneM;
    const float b2c  = b2[col];
    const int   rowB = m0 + (kHalf << 3);
#pragma unroll
    for (int r = 0; r < 8; ++r) {
        kern[(size_t)(rowB + r) * NK + col] = acc[r] + b2c;
    }
}

// ---------------------------------------------------------------------------
// Involution gather: one block per pixel; kern row (784 f32) staged in LDS;
// 256 threads == 256 channels -> fully coalesced x reads per tap.
// ---------------------------------------------------------------------------
__global__ __launch_bounds__(256) void involution_gather(
    const float* __restrict__ x, const float* __restrict__ kern,
    float* __restrict__ out)
{
    __shared__ float ksh[NK];
    const int pix = blockIdx.x;            // 0 .. 8191
    const int c   = threadIdx.x;           // 0 .. 255
    const int w   = pix & (HW - 1);
    const int h   = (pix >> 6) & (HW - 1);
    const int b   = pix >> 12;

    const float* kp = kern + (size_t)pix * NK;
    for (int i = threadIdx.x; i < NK; i += 256) ksh[i] = kp[i];
    __syncthreads();

    const int g = c & (G_NUM - 1);         // c % 16
    float acc = 0.0f;
#pragma unroll
    for (int kh = 0; kh < K_SZ; ++kh) {
        const int hh = h + kh - (K_SZ / 2);
        if ((unsigned)hh >= (unsigned)HW) continue;   // SAME zero padding
#pragma unroll
        for (int kw = 0; kw < K_SZ; ++kw) {
            const int ww = w + kw - (K_SZ / 2);
            if ((unsigned)ww >= (unsigned)HW) continue;
            const float xv = x[(((size_t)(b * HW + hh)) * HW + ww) * C_CH + c];
            acc = fmaf(ksh[(kh * K_SZ + kw) * G_NUM + g], xv, acc);
        }
    }
    out[(size_t)pix * C_CH + c] = acc;
}

// ---------------------------------------------------------------------------
extern "C" void kernel_launch(void* const* d_in, const int* in_sizes, int n_in,
                              void* d_out, int out_size, void* d_ws, size_t ws_size,
                              hipStream_t stream)
{
    const float* x     = (const float*)d_in[0];
    const float* w1    = (const float*)d_in[1];
    const float* b1    = (const float*)d_in[2];
    const float* gamma = (const float*)d_in[3];
    const float* beta  = (const float*)d_in[4];
    const float* mu    = (const float*)d_in[5];
    const float* var   = (const float*)d_in[6];
    const float* w2    = (const float*)d_in[7];
    const float* b2    = (const float*)d_in[8];

    float* out  = (float*)d_out;                       // (B,H,W,C)      = 2,097,152 f32
    float* kern = out + (size_t)M_TOT * C_CH;          // (B,H,W,KK,1,G) = 6,422,528 f32
    float* t    = (float*)d_ws;                        // (M_TOT, CR)    = 2 MB scratch

    // GEMM1: 2048 wave-tiles / 8 waves per block
    gemm1_bn_relu<<<256, 256, 0, stream>>>(x, w1, b1, gamma, beta, mu, var, t);
    // GEMM2: 25088 wave-tiles / 8 waves per block
    gemm2_kerngen<<<3136, 256, 0, stream>>>(t, w2, b2, kern);
    // Gather: one block per pixel
    involution_gather<<<M_TOT, 256, 0, stream>>>(x, kern, out);
}